// DisplacementNet_88313117540586
// MI455X (gfx1250) — compile-verified
//
#include <hip/hip_runtime.h>
#include <hip/hip_bf16.h>
#include <math.h>

// ---------------------------------------------------------------------------
// DisplacementNet forward for MI455X (gfx1250, wave32).
//   - kNN (k=12) via LDS-tiled brute force + register insertion sort
//   - Fourier features + projection / 4 graph-conv layers via fp32 WMMA
//     (V_WMMA_F32_16X16X4_F32), K zero-padded to multiples of 16
//   - GEMM staging: double-buffered LDS fed by GLOBAL_LOAD_ASYNC_TO_LDS_B128
//     (ASYNCcnt DMA, 4 b128/thread/chunk, overlapped with WMMA compute)
//   - weights pre-swizzled into WMMA fragment order (K-pair interleaved) so
//     every B fragment is a single aligned ds_load_b64
//   - layernorm + 192x3 output head, wave-per-point with shfl reductions
// ---------------------------------------------------------------------------

#define NPTS   32768
#define KNN    12
#define W_     192          // WIDTH
#define FEATK  96           // 83 padded to 96
#define MIXK   400          // 390 padded to 400
#define PROJ_ELEMS (FEATK * W_)        // 18432
#define LAY_ELEMS  (MIXK * W_)         // 76800

#define A_STRIDE  20        // floats per LDS A row (16 data + pad, 16B aligned)
#define WP_STRIDE 208       // float2 per LDS W pair-row (192 data + pad)

typedef __attribute__((ext_vector_type(2))) float v2f;
typedef __attribute__((ext_vector_type(8))) float v8f;

// ---- CDNA5 async global->LDS DMA (ASYNCcnt) --------------------------------
__device__ __forceinline__ void async_load_b128(unsigned lds_off, const void* gaddr)
{
    asm volatile("global_load_async_to_lds_b128 %0, %1, off"
                 :: "v"(lds_off), "v"((unsigned long long)(size_t)gaddr)
                 : "memory");
}
__device__ __forceinline__ void wait_async0()
{
    asm volatile("s_wait_asynccnt 0" ::: "memory");
}

// ---------------------------------------------------------------- weights ---
// Packed fragment order: Wp[kp*384 + 2*c + j] = W[2*kp + j][c]  (rows >= realK
// zero-padded).  One 16-K chunk = 8 pair-rows * 384 floats = 12288 B contiguous.
__global__ __launch_bounds__(256)
void pack_weights_kernel(const float* __restrict__ projW,
                         const float* __restrict__ linW,
                         float* __restrict__ wts)
{
    int gid = blockIdx.x * 256 + threadIdx.x;
    const int total = PROJ_ELEMS + 4 * LAY_ELEMS;
    if (gid >= total) return;
    if (gid < PROJ_ELEMS) {
        int kp = gid / 384, rem = gid % 384;
        int c = rem >> 1, j = rem & 1;
        int r = 2 * kp + j;
        wts[gid] = (r < 83) ? projW[r * W_ + c] : 0.0f;
    } else {
        int g  = gid - PROJ_ELEMS;
        int l  = g / LAY_ELEMS;
        int rr = g % LAY_ELEMS;
        int kp = rr / 384, rem = rr % 384;
        int c = rem >> 1, j = rem & 1;
        int r = 2 * kp + j;
        wts[gid] = (r < 390) ? linW[((size_t)l * 390 + r) * W_ + c] : 0.0f;
    }
}

// -------------------------------------------------------------------- kNN ---
__global__ __launch_bounds__(256)
void knn_kernel(const float* __restrict__ x, int* __restrict__ knn)
{
    __shared__ float sx[256], sy[256], sz[256];
    const int row = blockIdx.x * 256 + threadIdx.x;
    const float px = x[row * 3 + 0];
    const float py = x[row * 3 + 1];
    const float pz = x[row * 3 + 2];

    float bd[KNN];
    int   bi[KNN];
#pragma unroll
    for (int t = 0; t < KNN; ++t) { bd[t] = 3.0e38f; bi[t] = 0; }

    for (int tile = 0; tile < NPTS; tile += 256) {
        __syncthreads();
        int j = tile + threadIdx.x;
        sx[threadIdx.x] = x[j * 3 + 0];
        sy[threadIdx.x] = x[j * 3 + 1];
        sz[threadIdx.x] = x[j * 3 + 2];
        __syncthreads();
        for (int u = 0; u < 256; ++u) {
            int cj = tile + u;
            float dx = sx[u] - px, dy = sy[u] - py, dz = sz[u] - pz;
            float d = dx * dx + dy * dy + dz * dz;
            if (cj == row) continue;               // SELF_PENALTY equivalent
            if (d < bd[KNN - 1]) {
                float dd = d; int jj = cj;
#pragma unroll
                for (int t = 0; t < KNN; ++t) {
                    if (dd < bd[t]) {
                        float td = bd[t]; int ti = bi[t];
                        bd[t] = dd; bi[t] = jj;
                        dd = td; jj = ti;
                    }
                }
            }
        }
    }
#pragma unroll
    for (int t = 0; t < KNN; ++t) knn[row * KNN + t] = bi[t];
}

// --------------------------------------------------------------- features ---
// writes [sin(xB)(8) | cos(xB)(8) | x(3) | z(64) | zeros(13)] stride FEATK
__global__ __launch_bounds__(256)
void feat_kernel(const float* __restrict__ x, const float* __restrict__ z,
                 const float* __restrict__ Bf, float* __restrict__ feat)
{
    int gid = blockIdx.x * 256 + threadIdx.x;
    if (gid >= NPTS * FEATK) return;
    int i = gid / FEATK, c = gid % FEATK;
    float v = 0.0f;
    if (c < 16) {
        int f = c & 7;
        float xb = x[i * 3 + 0] * Bf[0 * 8 + f]
                 + x[i * 3 + 1] * Bf[1 * 8 + f]
                 + x[i * 3 + 2] * Bf[2 * 8 + f];
        v = (c < 8) ? sinf(xb) : cosf(xb);
    } else if (c < 19) {
        v = x[i * 3 + (c - 16)];
    } else if (c < 83) {
        v = z[c - 19];
    }
    feat[(size_t)i * FEATK + c] = v;
}

// ------------------------------------------------------- WMMA fp32 GEMM -----
// out(N x 192) = A(N x K) @ W(K x 192) + bias, optional silu.
// WmP is the pre-swizzled weight buffer (fragment order, see pack kernel).
// Block: 256 thr = 8 waves -> 64 rows x 192 cols. Wave: 16 rows x 96 cols.
__global__ __launch_bounds__(256)
void gemm_silu_kernel(const float* __restrict__ A, int K,
                      const float* __restrict__ WmP,
                      const float* __restrict__ bias,
                      float* __restrict__ out, int applySilu)
{
    __shared__ float alds[2][64 * A_STRIDE];       // 64 rows x 16 k (+pad)
    __shared__ float wlds[2][8 * WP_STRIDE * 2];   // 8 pair-rows x 192 fl2 (+pad)

    const int tid  = threadIdx.x;
    const int wave = tid >> 5;
    const int lane = tid & 31;
    const int half = lane >> 4;        // which 16-lane half
    const int l16  = lane & 15;
    const int rg   = wave >> 1;        // 0..3  row group
    const int cg   = wave & 1;         // 0..1  col group
    const int rowB = blockIdx.x * 64;
    const int row0 = rowB + rg * 16;
    const int col0 = cg * 96;

    // per-thread DMA tasks (4 x b128 per chunk): 1 for A, 3 for W
    const int arow = tid >> 2, aseg = tid & 3;     // A: 64 rows x 4 segs

    v8f acc[6] = {};                   // six 16x16 fp32 accumulators
    const int nChunks = K >> 4;

    auto stage = [&](int ch, int bufsel) {
        const int k0 = ch << 4;
        async_load_b128((unsigned)(size_t)&alds[bufsel][arow * A_STRIDE + aseg * 4],
                        &A[(size_t)(rowB + arow) * K + k0 + aseg * 4]);
        const float* wch = WmP + (size_t)ch * 3072;   // 8 pair-rows * 384 floats
#pragma unroll
        for (int j = 0; j < 3; ++j) {
            int w = tid + j * 256;                    // 0..767
            int p = w / 96, seg = w % 96;             // 96 b128 per pair-row
            async_load_b128((unsigned)(size_t)&wlds[bufsel][p * (WP_STRIDE * 2) + seg * 4],
                            &wch[p * 384 + seg * 4]);
        }
    };

    stage(0, 0);

    for (int ch = 0; ch < nChunks; ++ch) {
        wait_async0();                 // my DMAs done
        __syncthreads();               // everyone's DMAs done -> buf[cur] ready
        const int cur = ch & 1;

        if (ch + 1 < nChunks) stage(ch + 1, cur ^ 1);

        // preload all fragments of this chunk, then batch the 24 WMMAs
        v2f afr[4];
        v2f bfr[4][6];
#pragma unroll
        for (int k4 = 0; k4 < 4; ++k4) {
            const int kb = k4 * 4 + 2 * half;          // K offset of frag
            const int p  = k4 * 2 + half;              // pair-row index
            afr[k4] = *(const v2f*)&alds[cur][(rg * 16 + l16) * A_STRIDE + kb];
#pragma unroll
            for (int t = 0; t < 6; ++t) {
                const int c = col0 + t * 16 + l16;
                bfr[k4][t] = *(const v2f*)&wlds[cur][(p * WP_STRIDE + c) * 2];
            }
        }
#pragma unroll
        for (int k4 = 0; k4 < 4; ++k4) {
#pragma unroll
            for (int t = 0; t < 6; ++t) {
                acc[t] = __builtin_amdgcn_wmma_f32_16x16x4_f32(
                    false, afr[k4], false, bfr[k4][t], (short)0, acc[t],
                    false, false);
            }
        }
        __syncthreads();               // all reads of buf[cur] done before reuse
    }

#pragma unroll
    for (int t = 0; t < 6; ++t) {
        const int c = col0 + t * 16 + l16;
        const float bv = bias[c];
#pragma unroll
        for (int v = 0; v < 8; ++v) {
            const int r = row0 + v + 8 * half;
            float val = acc[t][v] + bv;
            if (applySilu) val = val / (1.0f + __expf(-val));
            out[(size_t)r * W_ + c] = val;
        }
    }
}

// ------------------------------------------------------------- mix build ----
// mix row: [h(192) | mean-nbr-h(192) | rel_mean(3) | rel_std(3) | zeros(10)]
__global__ __launch_bounds__(256)
void mix_kernel(const float* __restrict__ h, const float* __restrict__ x,
                const int* __restrict__ knn, float* __restrict__ mix)
{
    const int wave = threadIdx.x >> 5;
    const int lane = threadIdx.x & 31;
    const int i = blockIdx.x * 8 + wave;

    int nbr[KNN];
#pragma unroll
    for (int j = 0; j < KNN; ++j) nbr[j] = knn[i * KNN + j];

    float acc[6];
#pragma unroll
    for (int cc = 0; cc < 6; ++cc) acc[cc] = 0.0f;
#pragma unroll
    for (int j = 0; j < KNN; ++j) {
        const float* hr = h + (size_t)nbr[j] * W_;
#pragma unroll
        for (int cc = 0; cc < 6; ++cc) acc[cc] += hr[lane + 32 * cc];
    }

    const float* hi = h + (size_t)i * W_;
    float* mrow = mix + (size_t)i * MIXK;
#pragma unroll
    for (int cc = 0; cc < 6; ++cc) {
        int c = lane + 32 * cc;
        mrow[c]       = hi[c];
        mrow[W_ + c]  = acc[cc] * (1.0f / 12.0f);
    }

    // relative-coordinate stats (redundant per lane; loads are broadcast)
    const float xi0 = x[i * 3 + 0], xi1 = x[i * 3 + 1], xi2 = x[i * 3 + 2];
    float m0 = 0, m1 = 0, m2 = 0;
#pragma unroll
    for (int j = 0; j < KNN; ++j) {
        m0 += x[nbr[j] * 3 + 0] - xi0;
        m1 += x[nbr[j] * 3 + 1] - xi1;
        m2 += x[nbr[j] * 3 + 2] - xi2;
    }
    m0 *= (1.0f / 12.0f); m1 *= (1.0f / 12.0f); m2 *= (1.0f / 12.0f);
    float v0 = 0, v1 = 0, v2 = 0;
#pragma unroll
    for (int j = 0; j < KNN; ++j) {
        float d0 = (x[nbr[j] * 3 + 0] - xi0) - m0;
        float d1 = (x[nbr[j] * 3 + 1] - xi1) - m1;
        float d2 = (x[nbr[j] * 3 + 2] - xi2) - m2;
        v0 += d0 * d0; v1 += d1 * d1; v2 += d2 * d2;
    }
    if (lane == 0) {
        mrow[384] = m0; mrow[385] = m1; mrow[386] = m2;
        mrow[387] = sqrtf(v0 * (1.0f / 12.0f));
        mrow[388] = sqrtf(v1 * (1.0f / 12.0f));
        mrow[389] = sqrtf(v2 * (1.0f / 12.0f));
    }
    if (lane < 10) mrow[390 + lane] = 0.0f;   // K padding
}

// ---------------------------------------------------- layernorm + head ------
__global__ __launch_bounds__(256)
void final_kernel(const float* __restrict__ h,
                  const float* __restrict__ gamma, const float* __restrict__ beta,
                  const float* __restrict__ outW, const float* __restrict__ outb,
                  float* __restrict__ out)
{
    const int wave = threadIdx.x >> 5;
    const int lane = threadIdx.x & 31;
    const int i = blockIdx.x * 8 + wave;
    const float* hi = h + (size_t)i * W_;

    float vals[6];
    float s = 0.0f;
#pragma unroll
    for (int cc = 0; cc < 6; ++cc) { vals[cc] = hi[lane + 32 * cc]; s += vals[cc]; }
#pragma unroll
    for (int m = 16; m >= 1; m >>= 1) s += __shfl_xor(s, m, 32);
    const float mu = s * (1.0f / 192.0f);

    float q = 0.0f;
#pragma unroll
    for (int cc = 0; cc < 6; ++cc) { float d = vals[cc] - mu; q += d * d; }
#pragma unroll
    for (int m = 16; m >= 1; m >>= 1) q += __shfl_xor(q, m, 32);
    const float rstd = rsqrtf(q * (1.0f / 192.0f) + 1e-3f);

    float o0 = 0, o1 = 0, o2 = 0;
#pragma unroll
    for (int cc = 0; cc < 6; ++cc) {
        int c = lane + 32 * cc;
        float hn = (vals[cc] - mu) * rstd * gamma[c] + beta[c];
        o0 += hn * outW[c * 3 + 0];
        o1 += hn * outW[c * 3 + 1];
        o2 += hn * outW[c * 3 + 2];
    }
#pragma unroll
    for (int m = 16; m >= 1; m >>= 1) {
        o0 += __shfl_xor(o0, m, 32);
        o1 += __shfl_xor(o1, m, 32);
        o2 += __shfl_xor(o2, m, 32);
    }
    if (lane == 0) {
        out[i * 3 + 0] = o0 + outb[0];
        out[i * 3 + 1] = o1 + outb[1];
        out[i * 3 + 2] = o2 + outb[2];
    }
}

// ------------------------------------------------------------------ launch --
extern "C" void kernel_launch(void* const* d_in, const int* in_sizes, int n_in,
                              void* d_out, int out_size, void* d_ws, size_t ws_size,
                              hipStream_t stream)
{
    const float* x      = (const float*)d_in[0];   // (N,3)
    const float* z      = (const float*)d_in[1];   // (1,64)
    const float* Bf     = (const float*)d_in[2];   // (3,8)
    const float* projW  = (const float*)d_in[3];   // (83,192)
    const float* projB  = (const float*)d_in[4];   // (192,)
    const float* linW   = (const float*)d_in[5];   // (4,390,192)
    const float* linB   = (const float*)d_in[6];   // (4,192)
    const float* lnG    = (const float*)d_in[7];   // (192,)
    const float* lnB    = (const float*)d_in[8];   // (192,)
    const float* outW   = (const float*)d_in[9];   // (192,3)
    const float* outB   = (const float*)d_in[10];  // (3,)
    float* out          = (float*)d_out;           // (N,3)

    // workspace layout (bytes)
    char* ws = (char*)d_ws;
    int*   knn  = (int*)  (ws);                              // N*12 ints
    float* hbuf = (float*)(ws + 1572864);                    // N*192 f32
    float* mixb = (float*)(ws + 1572864 + 25165824);         // N*400 f32 (feat reuses it)
    float* wts  = (float*)(ws + 1572864 + 25165824 + 52428800); // packed weights

    const int wtotal = PROJ_ELEMS + 4 * LAY_ELEMS;
    pack_weights_kernel<<<(wtotal + 255) / 256, 256, 0, stream>>>(projW, linW, wts);
    knn_kernel<<<NPTS / 256, 256, 0, stream>>>(x, knn);
    feat_kernel<<<(NPTS * FEATK + 255) / 256, 256, 0, stream>>>(x, z, Bf, mixb);
    gemm_silu_kernel<<<NPTS / 64, 256, 0, stream>>>(mixb, FEATK, wts, projB, hbuf, 0);

    for (int l = 0; l < 4; ++l) {
        mix_kernel<<<NPTS / 8, 256, 0, stream>>>(hbuf, x, knn, mixb);
        gemm_silu_kernel<<<NPTS / 64, 256, 0, stream>>>(
            mixb, MIXK, wts + PROJ_ELEMS + (size_t)l * LAY_ELEMS,
            linB + (size_t)l * W_, hbuf, 1);
    }

    final_kernel<<<NPTS / 8, 256, 0, stream>>>(hbuf, lnG, lnB, outW, outB, out);
}